// Conv2D_55989193670925
// MI455X (gfx1250) — compile-verified
//
#include <hip/hip_runtime.h>
#include <cstdint>

// Problem constants (fixed by the reference)
#define N_BATCH 32
#define H 1024
#define W 1024
// Tiling
#define TH 32      // output rows per block
#define TW 128     // output cols per block
#define NBX (W / TW)   // 8
#define NBY (H / TH)   // 32
#define LROWS (TH + 2)
#define LSTR  136  // LDS row stride in floats: halo-L at idx 3, body 4..131 (16B aligned), halo-R at 132

typedef float vf4 __attribute__((ext_vector_type(4)));

// Async global->LDS fills (CDNA5 ASYNC path, tracked by ASYNCcnt).
// NT near (consumed once via LDS, keep out of WGP$), RT far (halo rows get L2 reuse).
__device__ __forceinline__ void async_ld_b128(uint32_t lds_byte_addr, const void* gaddr) {
    asm volatile("global_load_async_to_lds_b128 %0, %1, off th:TH_LOAD_NT_RT"
                 :: "v"(lds_byte_addr), "v"(gaddr) : "memory");
}
__device__ __forceinline__ void async_ld_b32(uint32_t lds_byte_addr, const void* gaddr) {
    asm volatile("global_load_async_to_lds_b32 %0, %1, off th:TH_LOAD_NT_RT"
                 :: "v"(lds_byte_addr), "v"(gaddr) : "memory");
}
__device__ __forceinline__ void wait_async0() {
#if __has_builtin(__builtin_amdgcn_s_wait_asynccnt)
    __builtin_amdgcn_s_wait_asynccnt(0);
#else
    asm volatile("s_wait_asynccnt 0" ::: "memory");
#endif
}

__global__ __launch_bounds__(256)
void conv3x3_async_tile(const float* __restrict__ X,
                        const float* __restrict__ Wt,
                        float* __restrict__ out) {
    __shared__ __align__(16) float tile[LROWS * LSTR];

    const int tid = threadIdx.x;
    const int bx = blockIdx.x, by = blockIdx.y;
    const int gx0 = bx * TW;
    const int gy0 = by * TH;
    const float* Xp = X + (size_t)blockIdx.z * H * W;

    // 1) Border blocks only: zero the tile so unloaded halo cells become the pad=1 zeros.
    //    (Interior blocks load every cell that is ever read, so they skip this + a barrier.)
    const bool border = (bx == 0) | (bx == NBX - 1) | (by == 0) | (by == NBY - 1);
    if (border) {
        vf4* t4 = (vf4*)tile;
        for (int i = tid; i < (LROWS * LSTR) / 4; i += 256)
            t4[i] = (vf4){0.f, 0.f, 0.f, 0.f};
        __syncthreads();
    }

    // 2) Async-stage the tile body: 34 rows x 32 float4 (cols gx0..gx0+127 -> LDS idx 4..131).
    for (int i = tid; i < LROWS * 32; i += 256) {
        const int r = i >> 5;
        const int k = i & 31;
        const int grow = gy0 - 1 + r;
        if ((unsigned)grow < (unsigned)H) {
            const uint32_t la = (uint32_t)(uintptr_t)&tile[r * LSTR + 4 + 4 * k];
            async_ld_b128(la, Xp + (size_t)grow * W + gx0 + 4 * k);
        }
    }
    // 3) Halo columns (global cols gx0-1 and gx0+128), one b32 per row where in-range.
    if (tid < 2 * LROWS) {
        const int side = (tid >= LROWS) ? 1 : 0;
        const int r    = side ? (tid - LROWS) : tid;
        const int grow = gy0 - 1 + r;
        const int gcol = side ? (gx0 + TW) : (gx0 - 1);
        if ((unsigned)grow < (unsigned)H && (unsigned)gcol < (unsigned)W) {
            const uint32_t la = (uint32_t)(uintptr_t)&tile[r * LSTR + (side ? 132 : 3)];
            async_ld_b32(la, Xp + (size_t)grow * W + gcol);
        }
    }
    wait_async0();      // drain this wave's ASYNCcnt (LDS writes landed)
    __syncthreads();    // make all waves' tiles visible

    // 4) Weights (uniform address -> scalar loads)
    const float w0 = Wt[0], w1 = Wt[1], w2 = Wt[2];
    const float w3 = Wt[3], w4 = Wt[4], w5 = Wt[5];
    const float w6 = Wt[6], w7 = Wt[7], w8 = Wt[8];

    // 5) Each thread: 1 output row x 16 cols, 144 FMAs from LDS.
    const int ty = tid >> 3;   // 0..31
    const int tx = tid & 7;    // 0..7 (16-col chunk)
    const float* r0 = &tile[ty * LSTR + 3 + 16 * tx];  // global row gy-1, col gx-1
    const float* r1 = r0 + LSTR;
    const float* r2 = r1 + LSTR;

    float o[16];
    #pragma unroll
    for (int i = 0; i < 16; ++i) {
        o[i] = w0 * r0[i] + w1 * r0[i + 1] + w2 * r0[i + 2]
             + w3 * r1[i] + w4 * r1[i + 1] + w5 * r1[i + 2]
             + w6 * r2[i] + w7 * r2[i + 1] + w8 * r2[i + 2];
    }

    // Streaming NT stores: output is never re-read; don't evict input lines from L2.
    vf4* op = (vf4*)(out + (size_t)blockIdx.z * H * W
                         + (size_t)(gy0 + ty) * W + gx0 + 16 * tx);
    #pragma unroll
    for (int i = 0; i < 4; ++i) {
        vf4 v = (vf4){o[4 * i], o[4 * i + 1], o[4 * i + 2], o[4 * i + 3]};
        __builtin_nontemporal_store(v, op + i);
    }
}

extern "C" void kernel_launch(void* const* d_in, const int* in_sizes, int n_in,
                              void* d_out, int out_size, void* d_ws, size_t ws_size,
                              hipStream_t stream) {
    const float* X  = (const float*)d_in[0];   // (32,1024,1024) f32
    const float* Wt = (const float*)d_in[1];   // (3,3) f32
    float* out = (float*)d_out;                // (32,1024,1024) f32

    dim3 grid(NBX, NBY, N_BATCH);              // 8 x 32 x 32
    conv3x3_async_tile<<<grid, dim3(256), 0, stream>>>(X, Wt, out);
}